// BilinearDecoder_72378788873076
// MI455X (gfx1250) — compile-verified
//
#include <hip/hip_runtime.h>
#include <hip/hip_bf16.h>

typedef __attribute__((ext_vector_type(2))) float v2f;
typedef __attribute__((ext_vector_type(8))) float v8f;

#define D 128

// ---------------------------------------------------------------------------
// Kernel 1: scored = embed @ S   (f32, exact, via V_WMMA_F32_16X16X4_F32)
// One wave computes a 16-row x 128-col slab; 8 waves/block -> 128 rows/block.
// S is staged in LDS in a K-paired layout so each B fragment (v2f = S[k][c],
// S[k+1][c]) is ONE contiguous ds_load_b64 straight into an even VGPR pair.
// ---------------------------------------------------------------------------
__global__ void __launch_bounds__(256)
scored_gemm_kernel(const float* __restrict__ embed,
                   const float* __restrict__ S,
                   float* __restrict__ scored,
                   int nrows) {
    // paired layout: sB[k2*256 + c*2 + 0] = S[2*k2][c]
    //                sB[k2*256 + c*2 + 1] = S[2*k2+1][c]      (64 KB total)
    __shared__ float sB[D * D];

    const int tid = threadIdx.x;
    // cooperative interleaving copy: each iter handles rows {2*k2, 2*k2+1},
    // cols c4..c4+3 -> two float4 LDS stores (32B contiguous).
    for (int idx = tid; idx < (D / 2) * (D / 4); idx += 256) {  // 2048 iters
        const int k2 = idx >> 5;            // 0..63
        const int c4 = (idx & 31) << 2;     // 0,4,...,124
        const float4 a = *(const float4*)&S[(2 * k2) * D + c4];
        const float4 b = *(const float4*)&S[(2 * k2 + 1) * D + c4];
        float4* dst = (float4*)&sB[k2 * 256 + c4 * 2];
        dst[0] = make_float4(a.x, b.x, a.y, b.y);
        dst[1] = make_float4(a.z, b.z, a.w, b.w);
    }
    __syncthreads();

    const int wave = tid >> 5;
    const int lane = tid & 31;
    const int r0   = (blockIdx.x * 8 + wave) * 16;
    if (r0 >= nrows) return;          // wave-uniform: EXEC all-ones inside

    // A 16x4 f32 fragment layout (ISA 7.12.2):
    //   lanes 0-15 : row M=lane,    v0=K+0, v1=K+1
    //   lanes16-31 : row M=lane-16, v0=K+2, v1=K+3
    const int mrow  = r0 + (lane & 15);
    const int khalf = (lane >> 4) << 1;       // 0 or 2
    const int bcol  = lane & 15;

    v8f acc[8];
#pragma unroll
    for (int n = 0; n < 8; ++n) acc[n] = (v8f){};

    const float* __restrict__ arow = embed + (size_t)mrow * D;

    for (int k = 0; k < D; k += 4) {
        const v2f a = *(const v2f*)(arow + k + khalf);     // global b64
        // B fragment base for this lane/k-step (paired layout):
        const float* bbase = sB + ((k >> 1) + (lane >> 4)) * 256 + bcol * 2;
#pragma unroll
        for (int n = 0; n < 8; ++n) {
            const v2f b = *(const v2f*)(bbase + n * 32);   // ds_load_b64
            acc[n] = __builtin_amdgcn_wmma_f32_16x16x4_f32(
                false, a, false, b, (short)0, acc[n], false, false);
        }
    }

    // C/D 16x16 f32 layout: VGPR i -> (row r0+i, col lane) for lanes 0-15,
    //                                  (row r0+8+i, col lane-16) for 16-31
    const int srow0 = r0 + ((lane >> 4) << 3);
    const int scol  = lane & 15;
#pragma unroll
    for (int n = 0; n < 8; ++n) {
#pragma unroll
        for (int i = 0; i < 8; ++i) {
            scored[(size_t)(srow0 + i) * D + n * 16 + scol] = acc[n][i];
        }
    }
}

// ---------------------------------------------------------------------------
// Kernel 2: per-edge scoring. One wave32 per edge; each lane owns 4 columns.
// All row gathers are 512B contiguous and lane-coalesced (float4/lane); the
// table (102.4MB) is L2-resident, so this stage is pure L2 bandwidth.
// ---------------------------------------------------------------------------
__device__ __forceinline__ float wave_reduce32(float v) {
#pragma unroll
    for (int m = 16; m >= 1; m >>= 1) v += __shfl_xor(v, m, 32);
    return v;
}

__device__ __forceinline__ float dot4(float4 a, float4 b) {
    return a.x * b.x + a.y * b.y + a.z * b.z + a.w * b.w;
}

__global__ void __launch_bounds__(256)
score_kernel(const float* __restrict__ embed,
             const float* __restrict__ scored,
             const int* __restrict__ h,
             const int* __restrict__ t,
             const int* __restrict__ n_s,
             float* __restrict__ out,
             int E) {
    const int gw   = (int)((blockIdx.x * blockDim.x + threadIdx.x) >> 5);
    const int lane = threadIdx.x & 31;
    if (gw >= E) return;

    const size_t hoff = (size_t)h[gw] * D + lane * 4;
    const size_t toff = (size_t)t[gw] * D + lane * 4;
    const float4 hs = *(const float4*)(scored + hoff);   // (embed[h] @ S) row
    const float4 te = *(const float4*)(embed + toff);

    float p = wave_reduce32(dot4(hs, te));
    if (lane == 0) out[gw] = p;                          // pos_score [E]

    float* __restrict__ neg = out + E + (size_t)gw * 16; // neg_score [E,16]

#pragma unroll
    for (int j = 0; j < 8; ++j) {                        // n_t half: hS . embed[n]
        const size_t off = (size_t)n_s[gw * 16 + j] * D + lane * 4;
        const float4 nt = *(const float4*)(embed + off);
        const float s = wave_reduce32(dot4(hs, nt));
        if (lane == 0) neg[j] = s;
    }
#pragma unroll
    for (int j = 0; j < 8; ++j) {                        // n_h half: (nS) . embed[t]
        const size_t off = (size_t)n_s[gw * 16 + 8 + j] * D + lane * 4;
        const float4 nh = *(const float4*)(scored + off);
        const float s = wave_reduce32(dot4(nh, te));
        if (lane == 0) neg[8 + j] = s;
    }
}

// ---------------------------------------------------------------------------
// Fallback (only if d_ws cannot hold the 102.4MB scored table): fused VALU
// path, one wave per edge, recomputes row @ S matvecs on the fly.
// ---------------------------------------------------------------------------
__device__ __forceinline__ void matvec_rowS(const float* __restrict__ row,
                                            const float* __restrict__ S,
                                            int lane, float r[4]) {
    r[0] = r[1] = r[2] = r[3] = 0.f;
    for (int k = 0; k < D; ++k) {
        const float a = row[k];                         // broadcast load
        const float4 s = *(const float4*)(S + (size_t)k * D + lane * 4);
        r[0] += a * s.x; r[1] += a * s.y; r[2] += a * s.z; r[3] += a * s.w;
    }
}

__global__ void __launch_bounds__(256)
fused_fallback_kernel(const float* __restrict__ embed,
                      const float* __restrict__ S,
                      const int* __restrict__ h,
                      const int* __restrict__ t,
                      const int* __restrict__ n_s,
                      float* __restrict__ out,
                      int E) {
    const int gw   = (int)((blockIdx.x * blockDim.x + threadIdx.x) >> 5);
    const int lane = threadIdx.x & 31;
    if (gw >= E) return;

    float hS[4];
    matvec_rowS(embed + (size_t)h[gw] * D, S, lane, hS);
    const float4 hs = make_float4(hS[0], hS[1], hS[2], hS[3]);
    const float4 te = *(const float4*)(embed + (size_t)t[gw] * D + lane * 4);

    float p = wave_reduce32(dot4(hs, te));
    if (lane == 0) out[gw] = p;

    float* __restrict__ neg = out + E + (size_t)gw * 16;
    for (int j = 0; j < 8; ++j) {
        const size_t off = (size_t)n_s[gw * 16 + j] * D + lane * 4;
        const float4 nt = *(const float4*)(embed + off);
        const float s = wave_reduce32(dot4(hs, nt));
        if (lane == 0) neg[j] = s;
    }
    for (int j = 0; j < 8; ++j) {
        float nS[4];
        matvec_rowS(embed + (size_t)n_s[gw * 16 + 8 + j] * D, S, lane, nS);
        const float4 nh = make_float4(nS[0], nS[1], nS[2], nS[3]);
        const float s = wave_reduce32(dot4(nh, te));
        if (lane == 0) neg[8 + j] = s;
    }
}

// ---------------------------------------------------------------------------
extern "C" void kernel_launch(void* const* d_in, const int* in_sizes, int n_in,
                              void* d_out, int out_size, void* d_ws, size_t ws_size,
                              hipStream_t stream) {
    const float* embed = (const float*)d_in[0];   // [N, 128] f32
    const float* S     = (const float*)d_in[1];   // [128, 128] f32
    const int*   h     = (const int*)d_in[2];     // [E, 1] i32
    const int*   t     = (const int*)d_in[3];     // [E, 1] i32
    const int*   n_s   = (const int*)d_in[4];     // [E, 16] i32
    float*       out   = (float*)d_out;           // [E] pos ++ [E,16] neg

    const int N = in_sizes[0] / D;                // 200000
    const int E = in_sizes[2];                    // 100000

    const size_t need = (size_t)N * D * sizeof(float);
    if (ws_size >= need) {
        float* scored = (float*)d_ws;
        const int waves  = (N + 15) / 16;
        const int blocks = (waves + 7) / 8;
        scored_gemm_kernel<<<blocks, 256, 0, stream>>>(embed, S, scored, N);
        const int eblocks = (E + 7) / 8;
        score_kernel<<<eblocks, 256, 0, stream>>>(embed, scored, h, t, n_s, out, E);
    } else {
        const int eblocks = (E + 7) / 8;
        fused_fallback_kernel<<<eblocks, 256, 0, stream>>>(embed, S, h, t, n_s, out, E);
    }
}